// FeaturesLoss_58033598103895
// MI455X (gfx1250) — compile-verified
//
#include <hip/hip_runtime.h>
#include <math.h>

typedef float v2f __attribute__((ext_vector_type(2)));
typedef float v8f __attribute__((ext_vector_type(8)));
typedef int   v4i __attribute__((vector_size(16)));   // matches builtin param pointee
typedef __attribute__((address_space(1))) v4i* gv4i_p;   // global (AS1)
typedef __attribute__((address_space(3))) v4i* lv4i_p;   // LDS (AS3)

#define N_PTS    8192
#define D_DIM    512
#define C_CLS    64
#define MARGIN_F 1.0f

#define TILE     128
#define NTB      (N_PTS / TILE)            // 64 tile rows/cols
#define NBLOCKS  (NTB * (NTB + 1) / 2)     // 2080 upper-triangular tiles
#define KBLK     32
#define LDSTR    36                        // padded LDS row stride (floats), 144B row pitch

// ---------------- CDNA5 async global->LDS copy (ASYNCcnt path) ----------------
__device__ __forceinline__ void async_copy16(const float* gsrc, float* ldst) {
#if __has_builtin(__builtin_amdgcn_global_load_async_to_lds_b128)
  __builtin_amdgcn_global_load_async_to_lds_b128(
      (gv4i_p)(v4i*)gsrc,   // global source (AS1), const dropped by C-cast
      (lv4i_p)(v4i*)ldst,   // LDS destination (AS3)
      0, 0);
#else
  unsigned int loff = (unsigned int)(unsigned long long)(uintptr_t)ldst;
  unsigned long long gaddr = (unsigned long long)(uintptr_t)gsrc;
  asm volatile("global_load_async_to_lds_b128 %0, %1, off"
               :: "v"(loff), "v"(gaddr) : "memory");
#endif
}

__device__ __forceinline__ void async_wait0() {
#if __has_builtin(__builtin_amdgcn_s_wait_asynccnt)
  __builtin_amdgcn_s_wait_asynccnt(0);
#else
  asm volatile("s_wait_asynccnt 0" ::: "memory");
#endif
}

// ---------------------------------------------------------------- argmax(labels)
__global__ void k_argmax(const float* __restrict__ labels, int* __restrict__ lbl) {
  int row = blockIdx.x * blockDim.x + threadIdx.x;
  if (row >= N_PTS) return;
  const float* p = labels + (size_t)row * C_CLS;
  float best = p[0]; int bi = 0;
  for (int c = 1; c < C_CLS; ++c) { float v = p[c]; if (v > best) { best = v; bi = c; } }
  lbl[row] = bi;
}

// ---------------------------------------------------------------- sq[i] = ||x_i||^2
__global__ void k_sqnorm(const float* __restrict__ X, float* __restrict__ sq) {
  int row  = (blockIdx.x * blockDim.x + threadIdx.x) >> 5;
  int lane = threadIdx.x & 31;
  if (row >= N_PTS) return;
  const float4* p = (const float4*)(X + (size_t)row * D_DIM);
  float s = 0.f;
  #pragma unroll
  for (int it = 0; it < D_DIM / 4; it += 32) {
    float4 v = p[it + lane];
    s += v.x * v.x + v.y * v.y + v.z * v.z + v.w * v.w;
  }
  #pragma unroll
  for (int off = 16; off > 0; off >>= 1) s += __shfl_xor(s, off, 32);
  if (lane == 0) sq[row] = s;
}

// ---------------------------------------------------------------- m1 = max class, m2 = next
__global__ void k_classinfo(const int* __restrict__ lbl, int* __restrict__ m1m2) {
  __shared__ unsigned long long sm[256];
  unsigned long long m = 0ull;
  for (int i = threadIdx.x; i < N_PTS; i += 256) m |= 1ull << (lbl[i] & 63);
  sm[threadIdx.x] = m;
  __syncthreads();
  for (int s = 128; s > 0; s >>= 1) {
    if (threadIdx.x < (unsigned)s) sm[threadIdx.x] |= sm[threadIdx.x + s];
    __syncthreads();
  }
  if (threadIdx.x == 0) {
    unsigned long long mm = sm[0];
    int m1 = -1, m2 = -1;
    for (int c = 63; c >= 0; --c)
      if ((mm >> c) & 1ull) { if (m1 < 0) m1 = c; else { m2 = c; break; } }
    m1m2[0] = m1; m1m2[1] = m2;
  }
}

// ---------------------------------------------------------------- main kernel
__global__ __launch_bounds__(256)
void k_gram(const float* __restrict__ X, const int* __restrict__ lbl,
            const float* __restrict__ sq, const int* __restrict__ m1m2,
            float* __restrict__ part) {
  __shared__ float sA[2][TILE * LDSTR];   // double-buffered A tile (rows)
  __shared__ float sB[2][TILE * LDSTR];   // double-buffered B tile (cols)
  __shared__ float sred[256];

  // block -> (br, bc) with br <= bc (upper triangle of 64x64 tile grid)
  int t = blockIdx.x, br = 0;
  while (t >= NTB - br) { t -= NTB - br; ++br; }
  const int bc = br + t;
  const int rowBase = br * TILE, colBase = bc * TILE;

  const int tid   = threadIdx.x;
  const int lane  = tid & 31;
  const int wave  = tid >> 5;
  const int waveM = wave >> 2;         // 0..1  -> 64 rows each
  const int waveN = wave & 3;          // 0..3  -> 32 cols each
  const int m16   = lane & 15;
  const int kOff  = (lane >> 4) << 1;  // lanes 16-31 hold K=2,3 (ISA 7.12.2)

  v8f acc[4][2];
  #pragma unroll
  for (int mt = 0; mt < 4; ++mt)
    #pragma unroll
    for (int nt = 0; nt < 2; ++nt) acc[mt][nt] = (v8f)0.0f;

  int aOff[4], bOff[2];
  #pragma unroll
  for (int mt = 0; mt < 4; ++mt) aOff[mt] = (waveM * 64 + mt * 16 + m16) * LDSTR + kOff;
  #pragma unroll
  for (int nt = 0; nt < 2; ++nt) bOff[nt] = (waveN * 32 + nt * 16 + m16) * LDSTR + kOff;

  const int ldRow = tid >> 3;         // 0..31 (rows per pass)
  const int ldCol = (tid & 7) << 2;   // 0,4,...,28

  // prologue: async-fill buffer 0 with K-block 0
  #pragma unroll
  for (int p = 0; p < 4; ++p) {
    int r = p * 32 + ldRow;
    async_copy16(X + (size_t)(rowBase + r) * D_DIM + ldCol, &sA[0][r * LDSTR + ldCol]);
    async_copy16(X + (size_t)(colBase + r) * D_DIM + ldCol, &sB[0][r * LDSTR + ldCol]);
  }
  async_wait0();
  __syncthreads();

  for (int k0 = 0; k0 < D_DIM; k0 += KBLK) {
    const int cur = (k0 >> 5) & 1;
    const float* curA = sA[cur];
    const float* curB = sB[cur];

    // prefetch next K-block into the other buffer (async, ASYNCcnt-tracked)
    if (k0 + KBLK < D_DIM) {
      float* nxtA = sA[cur ^ 1];
      float* nxtB = sB[cur ^ 1];
      const int kn = k0 + KBLK;
      #pragma unroll
      for (int p = 0; p < 4; ++p) {
        int r = p * 32 + ldRow;
        async_copy16(X + (size_t)(rowBase + r) * D_DIM + kn + ldCol, &nxtA[r * LDSTR + ldCol]);
        async_copy16(X + (size_t)(colBase + r) * D_DIM + kn + ldCol, &nxtB[r * LDSTR + ldCol]);
      }
    }

    // compute on current buffer, fragments software-pipelined one k-step ahead
    v2f af[2][4], bf[2][2];
    #pragma unroll
    for (int mt = 0; mt < 4; ++mt) af[0][mt] = *(const v2f*)(&curA[aOff[mt] + 0]);
    #pragma unroll
    for (int nt = 0; nt < 2; ++nt) bf[0][nt] = *(const v2f*)(&curB[bOff[nt] + 0]);

    #pragma unroll
    for (int kk = 0; kk < KBLK; kk += 4) {
      const int c = (kk >> 2) & 1;
      if (kk + 4 < KBLK) {
        #pragma unroll
        for (int mt = 0; mt < 4; ++mt) af[c ^ 1][mt] = *(const v2f*)(&curA[aOff[mt] + kk + 4]);
        #pragma unroll
        for (int nt = 0; nt < 2; ++nt) bf[c ^ 1][nt] = *(const v2f*)(&curB[bOff[nt] + kk + 4]);
      }
      #pragma unroll
      for (int mt = 0; mt < 4; ++mt)
        #pragma unroll
        for (int nt = 0; nt < 2; ++nt)
          acc[mt][nt] = __builtin_amdgcn_wmma_f32_16x16x4_f32(
              false, af[c][mt], false, bf[c][nt], (short)0, acc[mt][nt], false, false);
    }

    // my prefetch landed AND everyone is done reading `cur` => safe to proceed
    async_wait0();
    __syncthreads();
  }

  // fused epilogue: d2, same-class sum/count, cross-class hinge
  const int m1 = m1m2[0], m2 = m1m2[1];
  float sameS = 0.f, sameC = 0.f, hingeS = 0.f;
  const int hi8 = (lane >> 4) << 3;   // C/D layout: lanes 16-31 hold M = r + 8

  for (int nt = 0; nt < 2; ++nt) {
    const int j   = colBase + waveN * 32 + nt * 16 + m16;   // N = lane%16
    const int lj  = lbl[j];
    const float sqj = sq[j];
    for (int mt = 0; mt < 4; ++mt) {
      const int ibase = rowBase + waveM * 64 + mt * 16 + hi8;
      #pragma unroll
      for (int r = 0; r < 8; ++r) {
        const int i = ibase + r;
        if (i < j) {
          const float d2 = fmaxf(sq[i] + sqj - 2.0f * acc[mt][nt][r], 0.0f);
          const int li = lbl[i];
          if (li == lj) { sameS += d2; sameC += 1.0f; }
          if ((li == m2 && lj == m1) || (li == m1 && lj == m2)) {
            const float h = fmaxf(MARGIN_F - sqrtf(d2), 0.0f);
            hingeS += h * h;
          }
        }
      }
    }
  }

  // deterministic fixed-tree block reduction -> per-block partials
  float vals[3] = { sameS, sameC, hingeS };
  for (int p = 0; p < 3; ++p) {
    __syncthreads();
    sred[tid] = vals[p];
    __syncthreads();
    for (int s = 128; s > 0; s >>= 1) {
      if (tid < s) sred[tid] += sred[tid + s];
      __syncthreads();
    }
    if (tid == 0) part[blockIdx.x * 3 + p] = sred[0];
  }
}

// ---------------------------------------------------------------- deterministic finalize
__global__ void k_final(const float* __restrict__ part, float* __restrict__ out) {
  if (threadIdx.x != 0 || blockIdx.x != 0) return;
  float ss = 0.f, sc = 0.f, hs = 0.f;
  for (int b = 0; b < NBLOCKS; ++b) {
    ss += part[b * 3 + 0];
    sc += part[b * 3 + 1];
    hs += part[b * 3 + 2];
  }
  const float total_pairs = (float)((long long)N_PTS * (N_PTS - 1) / 2); // 33550336
  out[0] = ss / sc + hs / total_pairs;
}

// ----------------------------------------------------------------
extern "C" void kernel_launch(void* const* d_in, const int* in_sizes, int n_in,
                              void* d_out, int out_size, void* d_ws, size_t ws_size,
                              hipStream_t stream) {
  const float* X      = (const float*)d_in[0];   // [8192, 512] fp32
  const float* labels = (const float*)d_in[1];   // [8192, 64]  fp32
  float* out = (float*)d_out;

  // workspace layout
  int*   lbl  = (int*)d_ws;                             // 8192 ints   (32 KB)
  float* sq   = (float*)((char*)d_ws + 32768);          // 8192 floats (32 KB)
  int*   m1m2 = (int*)((char*)d_ws + 65536);            // 2 ints
  float* part = (float*)((char*)d_ws + 65536 + 256);    // 2080*3 floats

  k_argmax   <<<N_PTS / 256, 256, 0, stream>>>(labels, lbl);
  k_sqnorm   <<<N_PTS / 8,   256, 0, stream>>>(X, sq);
  k_classinfo<<<1,           256, 0, stream>>>(lbl, m1m2);
  k_gram     <<<NBLOCKS,     256, 0, stream>>>(X, lbl, sq, m1m2, part);
  k_final    <<<1,           1,   0, stream>>>(part, out);
}